// DistMaps_87497073754717
// MI455X (gfx1250) — compile-verified
//
#include <hip/hip_runtime.h>
#include <hip/hip_bf16.h>
#include <stdint.h>

// Problem constants (shapes fixed by setup_inputs()).
#define Dz 96
#define Hy 160
#define Wx 160
#define NB 2
#define NP 24
#define HW (Hy * Wx)                     // 25600
#define DHW ((size_t)Dz * Hy * Wx)       // 2457600
#define CHUNK 16384                      // floats per block = 64 KB LDS tile
#define CHUNKS_PER_B 150                 // DHW / CHUNK (exact)
#define NBLOCKS (NB * CHUNKS_PER_B)      // 300 (total out = 300*16384 = 4,915,200)
#define RAD 5.0f
#define RAD2 25.0f

typedef uint32_t u32x4 __attribute__((ext_vector_type(4)));
typedef uint32_t u32x8 __attribute__((ext_vector_type(8)));

__global__ __launch_bounds__(256)
void distmap_fill_kernel(const float* __restrict__ coords, float* __restrict__ out)
{
    __shared__ float zbuf[CHUNK];        // 64 KB of zeros, source for the TDM store
    const int tid = threadIdx.x;

    // ---- Phase 1: zero the LDS tile (vectorized, 16 float4 per thread) ----
    float4* z4 = (float4*)zbuf;
    #pragma unroll
    for (int i = 0; i < (CHUNK / 4) / 256; ++i) {
        z4[tid + i * 256] = make_float4(0.f, 0.f, 0.f, 0.f);
    }
    asm volatile("s_wait_dscnt 0x0" ::: "memory");  // LDS writes visible before TDM reads
    __syncthreads();

    const uint32_t chunk   = blockIdx.x;
    const size_t   chunkLo = (size_t)chunk * CHUNK;

    // ---- Phase 2: one TDM DMA per block: LDS zeros -> out[chunkLo, chunkLo+16384) ----
    // Scalar branch (readfirstlane) so only wave 0 issues: TDM ignores EXEC, a
    // predicated path would issue one DMA per wave.
    if (__builtin_amdgcn_readfirstlane(tid) < 32) {
        const uint64_t ga  = (uint64_t)(uintptr_t)(out + chunkLo);
        const uint32_t lds = (uint32_t)(uintptr_t)&zbuf[0];   // flat low 32 bits = LDS offset

        u32x4 g0;
        g0[0] = 1u;                                            // count=1, user descriptor
        g0[1] = lds;                                           // lds_addr (bytes)
        g0[2] = (uint32_t)(ga & 0xFFFFFFFFu);                  // global_addr[31:0]
        g0[3] = (uint32_t)((ga >> 32) & 0x1FFFFFFu)            // global_addr[56:32]
              | (0x2u << 30);                                  // type = 2 ("image")

        u32x8 g1;
        g1[0] = 2u << 16;                                      // wg_mask=0 | data_size=4B
        g1[1] = ((uint32_t)CHUNK & 0xFFFFu) << 16;             // atomic_addr=0 | tensor_dim0[15:0]
        g1[2] = ((uint32_t)CHUNK >> 16) | (1u << 16);          // tensor_dim0[31:16] | tensor_dim1=1
        g1[3] = ((uint32_t)CHUNK & 0xFFFFu) << 16;             // tensor_dim1 hi=0 | tile_dim0=16384
        g1[4] = 0u;                                            // tile_dim1/2 unused
        g1[5] = (uint32_t)CHUNK;                               // tensor_dim0_stride lo32
        g1[6] = 0u;                                            // stride0 hi | stride1 lo
        g1[7] = 0u;                                            // stride1 hi

        asm volatile("tensor_store_from_lds %0, %1"
                     :: "s"(g0), "s"(g1)
                     : "memory");
        __builtin_amdgcn_s_wait_tensorcnt(0);                  // DMA data written
    }
    __syncthreads();                                            // all waves see zeros landed

    // ---- Phase 3: sparse ball rasterization restricted to this block's chunk ----
    const int    b       = chunk / CHUNKS_PER_B;               // chunks never straddle batches
    const size_t chunkHi = chunkLo + CHUNK;
    const size_t bBase   = (size_t)b * DHW;

    for (int p = 0; p < NP; ++p) {
        const float zc = coords[(b * NP + p) * 3 + 0];
        const float yc = coords[(b * NP + p) * 3 + 1];
        const float xc = coords[(b * NP + p) * 3 + 2];

        int z0 = (int)ceilf(zc - RAD);  if (z0 < 0) z0 = 0;
        int z1 = (int)floorf(zc + RAD); if (z1 > Dz - 1) z1 = Dz - 1;
        int y0 = (int)ceilf(yc - RAD);  if (y0 < 0) y0 = 0;
        int y1 = (int)floorf(yc + RAD); if (y1 > Hy - 1) y1 = Hy - 1;
        int x0 = (int)ceilf(xc - RAD);  if (x0 < 0) x0 = 0;
        int x1 = (int)floorf(xc + RAD); if (x1 > Wx - 1) x1 = Wx - 1;
        if (z0 > z1 || y0 > y1 || x0 > x1) continue;

        // Flat-index range of the bounding box (monotone in d,h,w) vs this chunk.
        const size_t lo = bBase + (size_t)z0 * HW + (size_t)y0 * Wx + (size_t)x0;
        const size_t hi = bBase + (size_t)z1 * HW + (size_t)y1 * Wx + (size_t)x1;
        if (hi < chunkLo || lo >= chunkHi) continue;

        const int nx = x1 - x0 + 1, ny = y1 - y0 + 1, nz = z1 - z0 + 1;
        const int n  = nx * ny * nz;                            // <= 11^3 = 1331

        for (int i = tid; i < n; i += 256) {
            const int ix = i % nx;
            const int t  = i / nx;
            const int iy = t % ny;
            const int iz = t / ny;
            const int wd = x0 + ix, hh = y0 + iy, dd = z0 + iz;
            {
                #pragma clang fp contract(off)
                const float fx = (float)wd - xc;
                const float fy = (float)hh - yc;
                const float fz = (float)dd - zc;
                const float dist = fz * fz + fy * fy + fx * fx;  // matches JAX rounding
                if (dist <= RAD2) {
                    const size_t idx = bBase + (size_t)dd * HW + (size_t)hh * Wx + (size_t)wd;
                    if (idx >= chunkLo && idx < chunkHi) out[idx] = 1.0f;
                }
            }
        }
    }
}

extern "C" void kernel_launch(void* const* d_in, const int* in_sizes, int n_in,
                              void* d_out, int out_size, void* d_ws, size_t ws_size,
                              hipStream_t stream)
{
    (void)in_sizes; (void)n_in; (void)out_size; (void)d_ws; (void)ws_size;
    // d_in[0] = x (unused by the math, shape only), d_in[1] = coords (48 x 3 f32)
    const float* coords = (const float*)d_in[1];
    float* out = (float*)d_out;
    distmap_fill_kernel<<<NBLOCKS, 256, 0, stream>>>(coords, out);
}